// SpikeFP32ExtractLow5_76312978916072
// MI455X (gfx1250) — compile-verified
//
#include <hip/hip_runtime.h>
#include <stdint.h>

// ---------------------------------------------------------------------------
// SpikeFP32ExtractLow5: decode IEEE-754 bit-planes -> low 5 bits of integer
// Memory-bound (296 MB @ 23.3 TB/s ~= 12.7 us floor). CDNA5 path: TDM
// tensor_load_to_lds (double-buffered, LDS pad for conflict-free b128 reads),
// s_wait_tensorcnt, nontemporal streaming stores.
// ---------------------------------------------------------------------------

#define THREADS        256              // 8 wave32s
#define TILE_ELEMS     256              // elements per TDM tile (32 KB data)
#define NT             4                // tiles per block
#define ELEM_DW        32               // 32 dwords (128 B) per element
#define ELEM_STRIDE_DW 36               // padded LDS stride: 32 data + 4 pad dwords
#define TILE_DW        (TILE_ELEMS * ELEM_DW)         // 8192 dwords per tile
#define LDS_BUF_DW     (TILE_ELEMS * ELEM_STRIDE_DW)  // 9216 dwords per buffer

typedef __attribute__((ext_vector_type(4))) unsigned int tdm_u32x4;
typedef __attribute__((ext_vector_type(8))) int          tdm_i32x8;
typedef __attribute__((ext_vector_type(4))) int          tdm_i32x4;

#if defined(__AMDGCN__) && __has_builtin(__builtin_amdgcn_tensor_load_to_lds) && \
    __has_builtin(__builtin_amdgcn_s_wait_tensorcnt)
#define USE_TDM 1
#else
#define USE_TDM 0
#endif

// Low 32 bits of a flat shared-aperture address == LDS byte offset
// (CDNA5 ISA §10.2).
__device__ __forceinline__ unsigned lds_byte_offset(const void* p) {
  return (unsigned)(uintptr_t)p;
}

#if USE_TDM
// Build a D# (groups 0/1 per CDNA5 ISA) describing a 1-row
// tile of TILE_DW dwords starting at `tile_elem_base`, with LDS padding of
// 4 dwords every 32 dwords (pad_interval code 4, pad_amount code 3) so each
// 128 B element lands on a 144 B LDS stride (bank-conflict-free b128 reads).
__device__ __forceinline__ void tdm_issue_tile(const float* __restrict__ x,
                                               unsigned long long tile_elem_base,
                                               unsigned lds_off,
                                               unsigned long long total_dw) {
  const unsigned long long dw_start = tile_elem_base * (unsigned long long)ELEM_DW;
  const unsigned long long gaddr = (unsigned long long)(uintptr_t)x + dw_start * 4ull;
  unsigned long long rem = (dw_start < total_dw) ? (total_dw - dw_start) : 0ull;
  if (rem > 0xFFFFFFFFull) rem = 0xFFFFFFFFull;
  const unsigned dim0 = (unsigned)rem;   // OOB reads return zero (guarded at store)

  tdm_u32x4 g0;
  g0[0] = 1u;                                               // count=1, user mode
  g0[1] = lds_off;                                          // lds_addr (bytes)
  g0[2] = (unsigned)(gaddr & 0xFFFFFFFFull);                // global_addr[31:0]
  g0[3] = (unsigned)((gaddr >> 32) & 0x01FFFFFFull)         // global_addr[56:32]
          | (2u << 30);                                     // type = 2 ("image")

  tdm_i32x8 g1;
  g1[0] = (int)((2u << 16)        // data_size = 2 (4-byte elements)
                | (1u << 20)      // pad_enable
                | (4u << 22)      // pad_interval code 4 -> every 32 dwords
                | (3u << 25));    // pad_amount  code 3 -> 4 dwords
  g1[1] = (int)((dim0 & 0xFFFFu) << 16);            // tensor_dim0[15:0] @ [63:48]
  g1[2] = (int)((dim0 >> 16) | (1u << 16));         // tensor_dim0[31:16] | tensor_dim1=1
  g1[3] = (int)((unsigned)TILE_DW << 16);           // tile_dim0 = 8192 @ [127:112]
  g1[4] = 1;                                        // tile_dim1 = 1, tile_dim2 = 0
  g1[5] = (int)(unsigned)TILE_DW;                   // tensor_dim0_stride[31:0]
  g1[6] = 0;                                        // stride hi / dim1_stride lo
  g1[7] = 0;

  tdm_i32x4 g2 = {0, 0, 0, 0};   // dims 2..4 unused (tile_dim3/4 = 0)
  tdm_i32x4 g3 = {0, 0, 0, 0};
  tdm_i32x8 g4 = {0, 0, 0, 0, 0, 0, 0, 0};  // 6-arg toolchain form: zero-filled
  __builtin_amdgcn_tensor_load_to_lds(g0, g1, g2, g3, g4, /*cpol=*/0);
}
#endif

__global__ __launch_bounds__(THREADS) void
decode_low5_kernel(const float* __restrict__ x, float* __restrict__ out,
                   unsigned long long n_elems) {
  __shared__ unsigned int lds[2][LDS_BUF_DW];   // 2 x 36 KB double buffer
  const unsigned tid = threadIdx.x;
  const unsigned long long block_base =
      (unsigned long long)blockIdx.x * (unsigned long long)(NT * TILE_ELEMS);
  const unsigned long long total_dw = n_elems * (unsigned long long)ELEM_DW;

#if USE_TDM
  const bool wave0 = (tid >> 5) == 0;
  const unsigned lds_off0 = lds_byte_offset(&lds[0][0]);
  const unsigned lds_off1 = lds_byte_offset(&lds[1][0]);
  if (wave0) {
    tdm_issue_tile(x, block_base, lds_off0, total_dw);   // prime tile 0
  }
#endif

#pragma unroll
  for (int t = 0; t < NT; ++t) {
#if USE_TDM
    if (wave0) {
      if (t + 1 < NT) {
        // Overlap: issue next tile's DMA into the other buffer, then wait for
        // the *current* tile only (TENSORcnt <= 1; TDM completes in order).
        tdm_issue_tile(x, block_base + (unsigned long long)(t + 1) * TILE_ELEMS,
                       ((t + 1) & 1) ? lds_off1 : lds_off0, total_dw);
        __builtin_amdgcn_s_wait_tensorcnt(1);
      } else {
        __builtin_amdgcn_s_wait_tensorcnt(0);
      }
    }
    __syncthreads();   // publish TDM-written LDS to all 8 waves
#else
    {  // Fallback: cooperative copy, same padded LDS layout.
      const unsigned long long e =
          block_base + (unsigned long long)t * TILE_ELEMS + tid;
      unsigned int* dst = &lds[t & 1][tid * ELEM_STRIDE_DW];
      if (e < n_elems) {
        const uint4* src = reinterpret_cast<const uint4*>(x) + e * 8ull;
#pragma unroll
        for (int j = 0; j < 8; ++j) {
          *reinterpret_cast<uint4*>(dst + 4 * j) = src[j];
        }
      }
      __syncthreads();
    }
#endif

    // ---- compute: one element per thread, 8x ds_load_b128 (stride 144 B) ----
    const unsigned long long e =
        block_base + (unsigned long long)t * TILE_ELEMS + tid;
    const unsigned int* ep = &lds[t & 1][tid * ELEM_STRIDE_DW];
    unsigned w = 0u;
#pragma unroll
    for (int j = 0; j < 8; ++j) {
      // bits are exactly 0.0f/1.0f -> bit 23 of the raw pattern is the bit
      const uint4 q = *reinterpret_cast<const uint4*>(ep + 4 * j);
      w = (w << 4) | (((q.x >> 23) & 1u) << 3) | (((q.y >> 23) & 1u) << 2) |
          (((q.z >> 23) & 1u) << 1) | ((q.w >> 23) & 1u);
    }
    const int s = (int)((w >> 23) & 0xFFu) - 127;   // unbiased exponent
    unsigned low5 = 0u;                             // is_0..is_4 all false -> 0
    if (s >= 0 && s <= 4) {
      low5 = ((w & 0x7FFFFFu) | 0x800000u) >> (23 - s);  // (1<<s) | top-s mant
    }
    if (e < n_elems) {
      float* op = out + e * 5ull;                   // MSB-first (final_bits[::-1])
      __builtin_nontemporal_store((float)((low5 >> 4) & 1u), op + 0);
      __builtin_nontemporal_store((float)((low5 >> 3) & 1u), op + 1);
      __builtin_nontemporal_store((float)((low5 >> 2) & 1u), op + 2);
      __builtin_nontemporal_store((float)((low5 >> 1) & 1u), op + 3);
      __builtin_nontemporal_store((float)(low5 & 1u), op + 4);
    }
    __syncthreads();   // buffer t&1 free for reuse before tile t+2 is issued
  }
}

extern "C" void kernel_launch(void* const* d_in, const int* in_sizes, int n_in,
                              void* d_out, int out_size, void* d_ws, size_t ws_size,
                              hipStream_t stream) {
  const float* x = (const float*)d_in[0];
  float* out = (float*)d_out;
  const unsigned long long n_elems = (unsigned long long)in_sizes[0] / 32ull;   // 2,097,152
  const unsigned long long per_block = (unsigned long long)(NT * TILE_ELEMS);   // 1024
  const unsigned blocks = (unsigned)((n_elems + per_block - 1) / per_block);    // 2048
  decode_low5_kernel<<<dim3(blocks), dim3(THREADS), 0, stream>>>(x, out, n_elems);
  (void)d_ws; (void)ws_size; (void)out_size; (void)n_in;
}